// CPUForgetMult_48404281426010
// MI455X (gfx1250) — compile-verified
//
#include <hip/hip_runtime.h>
#include <hip/hip_bf16.h>
#include <stdint.h>

// h_t = f_t * h_{t-1} + (1 - f_t) * z_t  over T, independently per (b,h) channel.
// [T,B,H] = [1024,32,1024] fp32. Bandwidth-bound: 384 MB min traffic -> ~16.5us @ 23.3 TB/s.
// One thread per channel; async global->LDS ring buffer (depth 8, 8 timesteps/chunk)
// keeps ~14 MB of HBM traffic in flight despite only 1024 waves of parallelism.

#define BLOCK 256
#define CHUNK 8
#define DEPTH 8   // ring-buffer stages; 2 arrays * DEPTH * CHUNK * BLOCK * 4B = 128 KB LDS

#if defined(__AMDGCN__) && __has_builtin(__builtin_amdgcn_global_load_async_to_lds_b128)
#define USE_ASYNC 1
#else
#define USE_ASYNC 0
#endif

#if defined(__AMDGCN__)
#if __has_builtin(__builtin_amdgcn_s_wait_asynccnt)
#define WAIT_ASYNC(n) __builtin_amdgcn_s_wait_asynccnt(n)
#else
#define WAIT_ASYNC(n) asm volatile("s_wait_asynccnt %0" ::"n"(n) : "memory")
#endif
#endif

#if USE_ASYNC
typedef int v4i __attribute__((ext_vector_type(4)));
typedef __attribute__((address_space(1))) v4i gv4i;  // global int4
typedef __attribute__((address_space(3))) v4i lv4i;  // LDS int4

__device__ __forceinline__ gv4i* as_global(const void* p) {
  // global generic address == addrspace(1) address, value-preserving
  return (gv4i*)(uintptr_t)p;
}
__device__ __forceinline__ lv4i* as_lds(void* p) {
  // generic -> LDS addrspacecast
  return (lv4i*)p;
}

// Raw workgroup barrier WITHOUT the implicit memory-fence of __syncthreads():
// we do NOT want the compiler draining ASYNCcnt to 0 at every barrier — the
// explicit staged s_wait_asynccnt is the only async drain. LDS written by the
// async DMA is workgroup-visible once ASYNCcnt retires; the clobbers only stop
// compiler reordering of LDS accesses across the barrier.
__device__ __forceinline__ void block_barrier() {
  asm volatile("" ::: "memory");
  __builtin_amdgcn_s_barrier();
  asm volatile("" ::: "memory");
}
#endif

__global__ void __launch_bounds__(BLOCK) forgetmult_scan(
    const float* __restrict__ fp, const float* __restrict__ zp,
    float* __restrict__ hp, int T, int N) {
  const int tid = threadIdx.x;
  const int cb  = blockIdx.x * BLOCK;   // first channel of this block
  const int ch  = cb + tid;             // this thread's channel

#if USE_ASYNC
  __shared__ __align__(16) float sF[DEPTH][CHUNK * BLOCK];
  __shared__ __align__(16) float sZ[DEPTH][CHUNK * BLOCK];

  // Chunk slab copy: CHUNK rows x (BLOCK*4B = 1KB) per array.
  // Each 1KB row = 64 lanes * 16B; 256 threads cover 4 rows per sweep.
  const int sub = (tid & 63) << 2;  // element offset within row (16B granules)
  const int r0  = tid >> 6;         // starting row; uniform per wave32
  const int nchunk = T / CHUNK;

  auto issue_stage = [&](int c) {
    const int st = c & (DEPTH - 1);
#pragma unroll
    for (int r = 0; r < CHUNK; r += (BLOCK / 64)) {   // 2 sweeps -> 4 async instr/thread
      const int row  = r + r0;
      const size_t g = (size_t)(c * CHUNK + row) * (size_t)N + (size_t)(cb + sub);
      __builtin_amdgcn_global_load_async_to_lds_b128(
          as_global(fp + g), as_lds(&sF[st][row * BLOCK + sub]), 0, 0);
      __builtin_amdgcn_global_load_async_to_lds_b128(
          as_global(zp + g), as_lds(&sZ[st][row * BLOCK + sub]), 0, 0);
    }
  };

  const int pre = (nchunk < DEPTH) ? nchunk : DEPTH;
  for (int c = 0; c < pre; ++c) issue_stage(c);

  float hv = 0.0f;
  for (int c = 0; c < nchunk; ++c) {
    // Per-wave ASYNCcnt: 4 instructions per stage, completion is in-order.
    // Wait until only the stages issued *ahead* of c remain in flight.
    int ahead = nchunk - 1 - c;
    if (ahead > DEPTH - 1) ahead = DEPTH - 1;
    switch (ahead) {
      case 0:  WAIT_ASYNC(0);  break;
      case 1:  WAIT_ASYNC(4);  break;
      case 2:  WAIT_ASYNC(8);  break;
      case 3:  WAIT_ASYNC(12); break;
      case 4:  WAIT_ASYNC(16); break;
      case 5:  WAIT_ASYNC(20); break;
      case 6:  WAIT_ASYNC(24); break;
      default: WAIT_ASYNC(28); break;
    }
    block_barrier();  // all waves' stage-c loads now complete & visible

    const int st = c & (DEPTH - 1);
#pragma unroll
    for (int tt = 0; tt < CHUNK; ++tt) {
      const float fv = sF[st][tt * BLOCK + tid];
      const float zv = sZ[st][tt * BLOCK + tid];
      hv = __builtin_fmaf(fv, hv - zv, zv);  // f*h + (1-f)*z
      __builtin_nontemporal_store(hv, &hp[(size_t)(c * CHUNK + tt) * (size_t)N + ch]);
    }
    block_barrier();  // everyone done reading buffer st before it is refilled
    if (c + DEPTH < nchunk) issue_stage(c + DEPTH);
  }
#else
  // Fallback: coalesced sequential scan with one-step register prefetch.
  float hv = 0.0f;
  size_t idx = (size_t)ch;
  float fv = fp[idx], zv = zp[idx];
  for (int t = 0; t < T; ++t) {
    float fn = 0.0f, zn = 0.0f;
    if (t + 1 < T) { fn = fp[idx + (size_t)N]; zn = zp[idx + (size_t)N]; }
    hv = __builtin_fmaf(fv, hv - zv, zv);
    __builtin_nontemporal_store(hv, &hp[idx]);
    fv = fn; zv = zn; idx += (size_t)N;
  }
#endif
}

extern "C" void kernel_launch(void* const* d_in, const int* in_sizes, int n_in,
                              void* d_out, int out_size, void* d_ws, size_t ws_size,
                              hipStream_t stream) {
  (void)n_in; (void)d_ws; (void)ws_size; (void)out_size;
  const float* f = (const float*)d_in[0];  // forget gates
  const float* z = (const float*)d_in[1];  // inputs
  float* h = (float*)d_out;

  const int T = 1024;                 // time extent per reference
  const int N = in_sizes[0] / T;      // B*H = 32768 independent channels

  dim3 grid(N / BLOCK);
  forgetmult_scan<<<grid, BLOCK, 0, stream>>>(f, z, h, T, N);
}